// MGNNI_m_iter_52733608461004
// MI455X (gfx1250) — compile-verified
//
#include <hip/hip_runtime.h>
#include <hip/hip_bf16.h>
#include <math.h>

// MGNNI fixed-point solver for MI455X (gfx1250, wave32).
//   Z <- gamma * ( R @ G ) + Xnm,  R = (S^T)^2 Z  (node-major SpMM via L2 fp32 atomics)
//   G = F^T F / (||F^T F||_F + eps)  (symmetric => G^T == G)
// Dense GEMM uses V_WMMA_F32_16X16X4_F32 (fp32 matrix pipe) to preserve the
// 1e-6 convergence budget; all hot buffers (~103 MB) are L2-resident (192 MB).

#define FEAT 128
#define GAMMA_C 0.8f

typedef __attribute__((ext_vector_type(2))) float v2f;
typedef __attribute__((ext_vector_type(8))) float v8f;

// ---------------- FF = F^T F  and  sum(FF^2) ----------------
__global__ __launch_bounds__(256) void ffnorm_kernel(const float* __restrict__ F,
                                                     float* __restrict__ FF,
                                                     float* __restrict__ sumsq) {
    __shared__ float red[256];
    float local = 0.0f;
    for (int idx = threadIdx.x; idx < FEAT * FEAT; idx += 256) {
        const int j = idx >> 7;
        const int k = idx & 127;
        float s = 0.0f;
        #pragma unroll 8
        for (int i = 0; i < FEAT; ++i)
            s = fmaf(F[i * FEAT + j], F[i * FEAT + k], s);
        FF[idx] = s;
        local = fmaf(s, s, local);
    }
    red[threadIdx.x] = local;
    __syncthreads();
    for (int s = 128; s > 0; s >>= 1) {
        if ((int)threadIdx.x < s) red[threadIdx.x] += red[threadIdx.x + s];
        __syncthreads();
    }
    if (threadIdx.x == 0) *sumsq = red[0];
}

__global__ __launch_bounds__(256) void gnorm_kernel(float* __restrict__ G,
                                                    const float* __restrict__ sumsq) {
    const int idx = blockIdx.x * 256 + threadIdx.x;
    if (idx < FEAT * FEAT) {
        const float inv = 1.0f / (sqrtf(*sumsq) + 1e-12f);
        G[idx] *= inv;
    }
}

// ---------------- layout transforms ----------------
__global__ __launch_bounds__(256) void xpose_in_kernel(const float* __restrict__ X,
                                                       float* __restrict__ Xnm, int N) {
    const long long idx = (long long)blockIdx.x * 256 + threadIdx.x;
    if (idx >= (long long)N * FEAT) return;
    const int n = (int)(idx >> 7);
    const int m = (int)(idx & 127);
    Xnm[idx] = X[(long long)m * N + n];   // [M,N] -> node-major [N,M]
}

__global__ __launch_bounds__(256) void xpose_out_kernel(const float* __restrict__ P,
                                                        float* __restrict__ out, int N) {
    const long long idx = (long long)blockIdx.x * 256 + threadIdx.x;
    if (idx >= (long long)N * FEAT) return;
    const int m = (int)(idx / N);
    const int n = (int)(idx % N);
    out[idx] = P[(long long)n * FEAT + m]; // node-major [N,M] -> [M,N]
}

__global__ __launch_bounds__(256) void zero_kernel(float4* __restrict__ p, long long n4) {
    const long long i = (long long)blockIdx.x * 256 + threadIdx.x;
    if (i < n4) p[i] = make_float4(0.f, 0.f, 0.f, 0.f);
}

// ---------------- SpMM: Q[c] += w * P[r]  (one wave per edge) ----------------
__global__ __launch_bounds__(256) void spmm_kernel(const float* __restrict__ Pin,
                                                   float* __restrict__ Qout,
                                                   const int* __restrict__ rows,
                                                   const int* __restrict__ cols,
                                                   const float* __restrict__ w, int E) {
    const int wave = threadIdx.x >> 5;
    const int lane = threadIdx.x & 31;
    const int e = blockIdx.x * 8 + wave;
    if (e >= E) return;                      // wave-uniform guard
    const int r = rows[e];
    const int c = cols[e];
    const float wt = w[e];
    const float4 p = *(const float4*)(Pin + (long long)r * FEAT + lane * 4);
    float* q = Qout + (long long)c * FEAT + lane * 4;
    unsafeAtomicAdd(q + 0, wt * p.x);        // global_atomic_add_f32, resolves in L2
    unsafeAtomicAdd(q + 1, wt * p.y);
    unsafeAtomicAdd(q + 2, wt * p.z);
    unsafeAtomicAdd(q + 3, wt * p.w);
}

// ---------------- GEMM:  Z = gamma * (R @ G) + Xnm  via fp32 WMMA ----------------
// Block: 256 threads = 8 waves, 128 nodes per block; each wave owns 16 nodes and
// sweeps all 8 column tiles, reusing its register-cached A fragments.
__global__ __launch_bounds__(256) void gemm_wmma_kernel(const float* __restrict__ R,
                                                        const float* __restrict__ G,
                                                        const float* __restrict__ Xnm,
                                                        float* __restrict__ Z, int N) {
    __shared__ __align__(16) float sG[FEAT * FEAT];   // 64 KB of the 320 KB LDS
    const int tid = threadIdx.x;
    for (int i = tid * 4; i < FEAT * FEAT; i += 256 * 4)
        *(float4*)(sG + i) = *(const float4*)(G + i);
    __syncthreads();

    const int wave = tid >> 5;
    const int lane = tid & 31;
    const int half = lane >> 4;     // 0: lanes 0-15, 1: lanes 16-31
    const int l15  = lane & 15;
    const int nodeBase = blockIdx.x * 128 + wave * 16;
    if (nodeBase + 16 > N) return;  // wave-uniform (N % 16 == 0)

    // A fragments: 16x4 fp32 tiles. VGPR0 holds K=k0 (lanes 0-15) / K=k0+2
    // (lanes 16-31); VGPR1 holds K=k0+1 / K=k0+3. Cache all K=128.
    v2f aF[32];
    const float* rrow = R + (long long)(nodeBase + l15) * FEAT;
    #pragma unroll
    for (int ks = 0; ks < 32; ++ks) {
        const int k = ks * 4 + 2 * half;
        aF[ks].x = rrow[k];
        aF[ks].y = rrow[k + 1];
    }

    #pragma unroll
    for (int jt = 0; jt < 8; ++jt) {
        const int jBase = jt * 16;
        v8f acc = {0.f, 0.f, 0.f, 0.f, 0.f, 0.f, 0.f, 0.f};
        #pragma unroll
        for (int ks = 0; ks < 32; ++ks) {
            const int k = ks * 4 + 2 * half;
            v2f b;                               // B 4x16: rows striped across lanes
            b.x = sG[(k    ) * FEAT + jBase + l15];
            b.y = sG[(k + 1) * FEAT + jBase + l15];
            acc = __builtin_amdgcn_wmma_f32_16x16x4_f32(
                false, aF[ks], false, b, (short)0, acc, false, false);
        }
        #pragma unroll
        for (int v = 0; v < 8; ++v) {            // D: VGPR v -> row v (+8 for hi half)
            const int node = nodeBase + v + 8 * half;
            const int col  = jBase + l15;
            Z[(long long)node * FEAT + col] =
                GAMMA_C * acc[v] + Xnm[(long long)node * FEAT + col];
        }
    }
}

// ---------------- orchestration ----------------
extern "C" void kernel_launch(void* const* d_in, const int* in_sizes, int n_in,
                              void* d_out, int out_size, void* d_ws, size_t ws_size,
                              hipStream_t stream) {
    const float* X    = (const float*)d_in[0];   // [128, N]
    const float* F    = (const float*)d_in[1];   // [128, 128]
    const float* w    = (const float*)d_in[2];   // [E]
    const int*   rows = (const int*)d_in[3];     // [E]
    const int*   cols = (const int*)d_in[4];     // [E]
    const int N = in_sizes[0] / FEAT;            // 50000
    const int E = in_sizes[2];                   // 800000
    const long long MN = (long long)N * FEAT;

    float* ws    = (float*)d_ws;
    float* G     = ws;                       // 16384 floats
    float* sumsq = ws + FEAT * FEAT;         // 1 (padded to 128)
    float* Xnm   = ws + FEAT * FEAT + 128;   // N*128 node-major X
    float* P     = Xnm + MN;                 // Z (node-major)
    float* Q     = P + MN;                   // (S^T) Z
    float* Rb    = Q + MN;                   // (S^T)^2 Z

    const int gridT = (int)((MN + 255) / 256);
    const int gridZ = (int)((MN / 4 + 255) / 256);
    const int gridE = (E + 7) / 8;
    const int gridG = (N + 127) / 128;

    // G = F^T F / (||F^T F||_F + eps)
    ffnorm_kernel<<<1, 256, 0, stream>>>(F, G, sumsq);
    gnorm_kernel<<<64, 256, 0, stream>>>(G, sumsq);

    // Xnm = X^T ; Z1 = f(0) = X
    xpose_in_kernel<<<gridT, 256, 0, stream>>>(X, Xnm, N);
    hipMemcpyAsync(P, Xnm, MN * sizeof(float), hipMemcpyDeviceToDevice, stream);

    // 15 fixed-point iterations (contraction: running to max_iter only tightens
    // the fixed point vs. the reference's early exit)
    for (int it = 0; it < 15; ++it) {
        zero_kernel<<<gridZ, 256, 0, stream>>>((float4*)Q, MN / 4);
        spmm_kernel<<<gridE, 256, 0, stream>>>(P, Q, rows, cols, w, E);
        zero_kernel<<<gridZ, 256, 0, stream>>>((float4*)Rb, MN / 4);
        spmm_kernel<<<gridE, 256, 0, stream>>>(Q, Rb, rows, cols, w, E);
        gemm_wmma_kernel<<<gridG, 256, 0, stream>>>(Rb, G, Xnm, P, N);
    }

    xpose_out_kernel<<<gridT, 256, 0, stream>>>(P, (float*)d_out, N);
}